// Quantizer_74826920231488
// MI455X (gfx1250) — compile-verified
//
#include <hip/hip_runtime.h>

typedef __attribute__((ext_vector_type(2))) float v2f;
typedef __attribute__((ext_vector_type(4))) float v4f;
typedef __attribute__((ext_vector_type(8))) float v8f;

#define VQ_B 32
#define VQ_T 4096
#define VQ_D 64
#define VQ_K 512
#define VQ_N (VQ_B * VQ_T)            // 131072 rows
#define ROWS_PER_WG 256               // 8 waves x 32 rows
#define NUM_WG (VQ_N / ROWS_PER_WG)   // 512
#define CSTRIDE 68                    // dwords per codebook column (64 + 4 pad)
                                      // 68 % 64 == 4 -> wave halves on disjoint bank sets

// workspace float layout
#define WS_PARTIAL 0                  // NUM_WG floats
#define WS_CNORM   NUM_WG             // VQ_K floats

// ---------------------------------------------------------------------------
// Kernel 1: column norms ||E[:,k]||^2
// ---------------------------------------------------------------------------
__global__ void vq_colnorm(const float* __restrict__ E, float* __restrict__ cn) {
    int k = blockIdx.x * blockDim.x + threadIdx.x;
    if (k >= VQ_K) return;
    float s = 0.f;
#pragma unroll
    for (int d = 0; d < VQ_D; ++d) {
        float e = E[d * VQ_K + k];
        s += e * e;
    }
    cn[k] = s;
}

// ---------------------------------------------------------------------------
// Kernel 2: main — WMMA distance matmul + argmin + gather + loss partials
// WG = 256 threads = 8 waves; each wave owns 32 rows x all 512 columns.
// Codebook in LDS, column-major padded: E[d][c] at dword c*CSTRIDE + d.
//   -> B fragments: one aligned ds_load_b64; per-kk offsets are 16B immediates.
//   -> q gather: one aligned ds_load_b128 per lane.
// ---------------------------------------------------------------------------
__global__ void __launch_bounds__(256)
vq_main(const float* __restrict__ X, const float* __restrict__ E,
        const float* __restrict__ cnorm, float* __restrict__ q,
        float* __restrict__ partial) {
    extern __shared__ float smem[];
    float* sE = smem;                    // VQ_K * CSTRIDE dwords
    float* sN = smem + VQ_K * CSTRIDE;   // 512
    float* sR = sN + VQ_K;               // 8

    const int tid = threadIdx.x;

    // Stage codebook into column-major padded LDS (coalesced global reads).
#pragma unroll
    for (int i = 0; i < (VQ_D * VQ_K) / 256; ++i) {
        int idx = i * 256 + tid;
        int d = idx >> 9;
        int c = idx & (VQ_K - 1);
        sE[c * CSTRIDE + d] = E[idx];
    }
    sN[tid]       = cnorm[tid];
    sN[tid + 256] = cnorm[tid + 256];
    __syncthreads();

    const int wave = tid >> 5;
    const int lane = tid & 31;
    const int g    = lane & 15;   // position within 16-lane group
    const int half = lane >> 4;   // 0: lanes 0-15, 1: lanes 16-31
    const int rowBase = blockIdx.x * ROWS_PER_WG + wave * 32;

    // A fragments for two 16-row M-tiles (V_WMMA_F32_16X16X4_F32 A layout:
    // lane l holds row l%16; half 0 -> K={0,1}+4kk, half 1 -> K={2,3}+4kk).
    v2f a0[16], a1[16];
    const float* xr0 = X + (size_t)(rowBase + g) * VQ_D + half * 2;
    const float* xr1 = xr0 + 16 * VQ_D;
    float rn0 = 0.f, rn1 = 0.f;
#pragma unroll
    for (int kk = 0; kk < 16; ++kk) {
        a0[kk] = *(const v2f*)(xr0 + kk * 4);
        a1[kk] = *(const v2f*)(xr1 + kk * 4);
        rn0 += a0[kk].x * a0[kk].x + a0[kk].y * a0[kk].y;
        rn1 += a1[kk].x * a1[kk].x + a1[kk].y * a1[kk].y;
    }
    rn0 += __shfl_xor(rn0, 16, 32);   // ||x_row||^2 replicated at lanes g, g+16
    rn1 += __shfl_xor(rn1, 16, 32);

    float minv[2][8];
    int   mini[2][8];
#pragma unroll
    for (int s = 0; s < 2; ++s)
#pragma unroll
        for (int j = 0; j < 8; ++j) { minv[s][j] = 3.4e38f; mini[s][j] = 0; }

    // B fragment lane base: column (n0+g), element kk*4 + half*2.
    const float* bbase = sE + g * CSTRIDE + half * 2;

    for (int n0 = 0; n0 < VQ_K; n0 += 16) {
        const float* bt = bbase + n0 * CSTRIDE;   // one add per tile
        v8f c00 = {}, c01 = {}, c10 = {}, c11 = {};
#pragma unroll
        for (int kk = 0; kk < 16; kk += 2) {
            v2f b0 = *(const v2f*)(bt + kk * 4);        // ds_load_b64, imm offset
            c00 = __builtin_amdgcn_wmma_f32_16x16x4_f32(
                      false, a0[kk], false, b0, (short)0, c00, false, false);
            c10 = __builtin_amdgcn_wmma_f32_16x16x4_f32(
                      false, a1[kk], false, b0, (short)0, c10, false, false);
            v2f b1 = *(const v2f*)(bt + kk * 4 + 4);    // ds_load_b64, imm offset
            c01 = __builtin_amdgcn_wmma_f32_16x16x4_f32(
                      false, a0[kk + 1], false, b1, (short)0, c01, false, false);
            c11 = __builtin_amdgcn_wmma_f32_16x16x4_f32(
                      false, a1[kk + 1], false, b1, (short)0, c11, false, false);
        }
        // distance minus the row-constant ||x||^2:  cn - 2*dot
        float cn  = sN[n0 + g];
        int   col = n0 + g;
#pragma unroll
        for (int j = 0; j < 8; ++j) {
            float v0 = fmaf(-2.f, c00[j] + c01[j], cn);
            if (v0 < minv[0][j]) { minv[0][j] = v0; mini[0][j] = col; }
            float v1 = fmaf(-2.f, c10[j] + c11[j], cn);
            if (v1 < minv[1][j]) { minv[1][j] = v1; mini[1][j] = col; }
        }
    }

    // Butterfly min-reduction across each 16-lane group (tie -> smaller index,
    // matching jnp.argmax first-occurrence). xor<16 stays within the group.
#pragma unroll
    for (int off = 1; off < 16; off <<= 1) {
#pragma unroll
        for (int s = 0; s < 2; ++s)
#pragma unroll
            for (int j = 0; j < 8; ++j) {
                float ov = __shfl_xor(minv[s][j], off, 32);
                int   oi = __shfl_xor(mini[s][j], off, 32);
                if (ov < minv[s][j] || (ov == minv[s][j] && oi < mini[s][j])) {
                    minv[s][j] = ov; mini[s][j] = oi;
                }
            }
    }
    // half 0 lanes hold rows +0..7 (per set), half 1 lanes rows +8..15.

    // Gather q rows: column idx is contiguous in LDS; one b128 per lane
    // (byte addr = idx*272 + 16*g, 16B aligned), then a 16B global store.
#pragma unroll
    for (int s = 0; s < 2; ++s) {
#pragma unroll
        for (int j = 0; j < 8; ++j) {
            int row = rowBase + s * 16 + half * 8 + j;
            int idx = mini[s][j];
            v4f qv = *(const v4f*)(sE + idx * CSTRIDE + g * 4);
            *(v4f*)(q + (size_t)row * VQ_D + g * 4) = qv;
        }
    }

    // Per-row squared error = minv + rowNorm; sum the wave's 32 rows.
    float smin = 0.f;
#pragma unroll
    for (int s = 0; s < 2; ++s)
#pragma unroll
        for (int j = 0; j < 8; ++j) smin += minv[s][j];
    float t = ((g == 0) ? smin : 0.f) + ((half == 0) ? (rn0 + rn1) : 0.f);
#pragma unroll
    for (int off = 1; off < 32; off <<= 1) t += __shfl_xor(t, off, 32);
    if (lane == 0) sR[wave] = t;
    __syncthreads();
    if (tid == 0) {
        float s = 0.f;
#pragma unroll
        for (int w = 0; w < 8; ++w) s += sR[w];
        partial[blockIdx.x] = s;
    }
}

// ---------------------------------------------------------------------------
// Kernel 3: deterministic reduction of per-WG loss partials (512 values)
// ---------------------------------------------------------------------------
__global__ void vq_reduce(const float* __restrict__ partial, float* __restrict__ lossOut) {
    __shared__ float s[256];
    int t = threadIdx.x;
    float v = partial[t] + partial[t + 256];
    s[t] = v;
    __syncthreads();
    for (int o = 128; o > 0; o >>= 1) {
        if (t < o) s[t] += s[t + o];
        __syncthreads();
    }
    if (t == 0)
        lossOut[0] = s[0] * (2.0f / (float)((size_t)VQ_N * VQ_D));  // commit + codebook
}

// ---------------------------------------------------------------------------
extern "C" void kernel_launch(void* const* d_in, const int* in_sizes, int n_in,
                              void* d_out, int out_size, void* d_ws, size_t ws_size,
                              hipStream_t stream) {
    const float* X = (const float*)d_in[0];   // [B,T,D] f32
    const float* E = (const float*)d_in[1];   // [D,K]   f32
    float* out     = (float*)d_out;           // q (N*D floats) then loss (1 float)
    float* ws      = (float*)d_ws;
    float* partial = ws + WS_PARTIAL;
    float* cnorm   = ws + WS_CNORM;

    vq_colnorm<<<(VQ_K + 255) / 256, 256, 0, stream>>>(E, cnorm);

    size_t shbytes = (size_t)(VQ_K * CSTRIDE + VQ_K + 8) * sizeof(float);  // ~138 KB
    vq_main<<<NUM_WG, 256, shbytes, stream>>>(X, E, cnorm, out, partial);

    vq_reduce<<<1, 256, 0, stream>>>(partial, out + (size_t)VQ_N * VQ_D);
}